// CTGAP_89240830476826
// MI455X (gfx1250) — compile-verified
//
#include <hip/hip_runtime.h>
#include <math.h>

#define B_  16
#define H_  512
#define N_  2048
#define DN_ 256
#define G_  64
#define C_  (B_ * G_)   // 1024 columns of U/P

typedef __attribute__((ext_vector_type(2))) float v2f;
typedef __attribute__((ext_vector_type(8))) float v8f;

// ---------------- small projection kernels ----------------

__global__ void q_kern(const float* __restrict__ h_t, const float* __restrict__ Wq,
                       const float* __restrict__ bq, float* __restrict__ q) {
  int i = blockIdx.x * blockDim.x + threadIdx.x;     // B*G = 1024
  if (i >= B_ * G_) return;
  int b = i >> 6, g = i & 63;
  const float* hr = h_t + b * H_;
  const float* wr = Wq + g * H_;
  float s = bq[g];
  for (int h = 0; h < H_; ++h) s += hr[h] * wr[h];
  q[i] = s;
}

__global__ void kv_kern(const float* __restrict__ nr,
                        const float* __restrict__ Wk, const float* __restrict__ bk,
                        const float* __restrict__ Wv, const float* __restrict__ bv,
                        float* __restrict__ k, float* __restrict__ v) {
  int i = blockIdx.x * blockDim.x + threadIdx.x;     // N*G = 131072
  if (i >= N_ * G_) return;
  int n = i >> 6, g = i & 63;
  const float* x  = nr + n * DN_;
  const float* wk = Wk + g * DN_;
  const float* wv = Wv + g * DN_;
  float sk = bk[g], sv = bv[g];
  for (int d = 0; d < DN_; ++d) { float xv = x[d]; sk += xv * wk[d]; sv += xv * wv[d]; }
  k[i] = sk; v[i] = sv;
}

__global__ void rowsum_kern(const float* __restrict__ adj, float* __restrict__ rowsum) {
  __shared__ float red[256];
  int n = blockIdx.x;
  float s = 0.f;
  for (int m = threadIdx.x; m < N_; m += 256) s += adj[n * N_ + m];
  red[threadIdx.x] = s; __syncthreads();
  for (int off = 128; off > 0; off >>= 1) {
    if (threadIdx.x < off) red[threadIdx.x] += red[threadIdx.x + off];
    __syncthreads();
  }
  if (threadIdx.x == 0) rowsum[n] = red[0];
}

// ---------------- attention ----------------

__global__ void logits_kern(const float* __restrict__ q, const float* __restrict__ k,
                            float* __restrict__ logits) {
  int i = blockIdx.x * blockDim.x + threadIdx.x;     // B*N
  if (i >= B_ * N_) return;
  int b = i / N_, n = i % N_;
  const float* qr = q + b * G_;
  const float* kr = k + n * G_;
  float s = 0.f;
#pragma unroll
  for (int g = 0; g < G_; ++g) s += qr[g] * kr[g];
  logits[i] = s * 0.125f;                            // 1/sqrt(64)
}

__global__ void smstats_kern(const float* __restrict__ logits,
                             float* __restrict__ rowmax, float* __restrict__ rowexp) {
  __shared__ float red[256];
  int b = blockIdx.x;
  const float* l = logits + b * N_;
  float mx = -3.4e38f;
  for (int n = threadIdx.x; n < N_; n += 256) mx = fmaxf(mx, l[n]);
  red[threadIdx.x] = mx; __syncthreads();
  for (int off = 128; off > 0; off >>= 1) {
    if (threadIdx.x < off) red[threadIdx.x] = fmaxf(red[threadIdx.x], red[threadIdx.x + off]);
    __syncthreads();
  }
  mx = red[0]; __syncthreads();
  float s = 0.f;
  for (int n = threadIdx.x; n < N_; n += 256) s += expf(l[n] - mx);
  red[threadIdx.x] = s; __syncthreads();
  for (int off = 128; off > 0; off >>= 1) {
    if (threadIdx.x < off) red[threadIdx.x] += red[threadIdx.x + off];
    __syncthreads();
  }
  if (threadIdx.x == 0) { rowmax[b] = mx; rowexp[b] = red[0]; }
}

__global__ void alpha_kern(const float* __restrict__ logits, const float* __restrict__ rowmax,
                           const float* __restrict__ rowexp, const float* __restrict__ rowsum,
                           float* __restrict__ alpha, float* __restrict__ dinv) {
  int i = blockIdx.x * blockDim.x + threadIdx.x;     // B*N
  if (i >= B_ * N_) return;
  int b = i / N_, n = i % N_;
  float a = expf(logits[i] - rowmax[b]) / rowexp[b];
  alpha[i] = a;
  dinv[i] = rsqrtf(a * rowsum[n] + 1.0f + 1e-8f);    // (deg + eps)^-1/2, deg = a*rowsum + 1
}

__global__ void u_kern(const float* __restrict__ dinv, const float* __restrict__ v,
                       float* __restrict__ U) {
  int i = blockIdx.x * blockDim.x + threadIdx.x;     // N*C
  if (i >= N_ * C_) return;
  int m = i >> 10, c = i & (C_ - 1);
  int b = c >> 6, g = c & 63;
  U[i] = dinv[b * N_ + m] * v[m * G_ + g];
}

__global__ void rv_kern(const float* __restrict__ alpha, const float* __restrict__ v,
                        float* __restrict__ rv) {
  __shared__ float red[256];
  int b = blockIdx.x >> 6, g = blockIdx.x & 63;
  float s = 0.f;
  for (int n = threadIdx.x; n < N_; n += 256) s += alpha[b * N_ + n] * v[n * G_ + g];
  red[threadIdx.x] = s; __syncthreads();
  for (int off = 128; off > 0; off >>= 1) {
    if (threadIdx.x < off) red[threadIdx.x] += red[threadIdx.x + off];
    __syncthreads();
  }
  if (threadIdx.x == 0) rv[b * G_ + g] = red[0];
}

// ---------------- main GEMM: P = adj[2048x2048] @ U[2048x1024], f32 WMMA ----------------
// Block tile 256x64, K-chunk 32. 8 waves, each wave owns 32 rows x 64 cols
// = 8 v8f accumulators. All LDS fragment reads are single aligned b64 loads
// (compiler fuses them into ds_load_2addr_b64 pairs). Global->LDS copy is a
// short-liveness phase (no cross-compute staging => no spills); latency for
// the next chunk is hidden with global_prefetch (__builtin_prefetch).

#define TM 256
#define TN 64
#define TK 32
#define AS_STRIDE 36            // even, x4 -> float4 stores + conflict-free b64 reads
#define BP_STRIDE 80            // float2 units; 160 words = 32 mod 64 banks

__global__ __launch_bounds__(256) void gemm_adj_u(const float* __restrict__ adj,
                                                  const float* __restrict__ U,
                                                  float* __restrict__ P) {
  __shared__ __align__(16) float  As[TM * AS_STRIDE];       // 36 KB
  __shared__ __align__(16) float2 Bp[(TK / 2) * BP_STRIDE]; // 10 KB: Bp[kp][n]={B[2kp][n],B[2kp+1][n]}

  const int tid  = threadIdx.x;
  const int wave = tid >> 5;
  const int lane = tid & 31;
  const int m16  = lane & 15;
  const int half = lane >> 4;
  const int rowBlk  = blockIdx.y * TM;
  const int colBlk  = blockIdx.x * TN;
  const int rowBase = wave * 32;

  // per-thread tile coordinates for the copy phases
  const int ar[8] = { tid >> 3, (256 + tid) >> 3, (512 + tid) >> 3, (768 + tid) >> 3,
                      (1024 + tid) >> 3, (1280 + tid) >> 3, (1536 + tid) >> 3, (1792 + tid) >> 3 };
  const int ac4 = (tid & 7) * 4;                    // same column group for all j
  const int br[2] = { tid >> 4, (256 + tid) >> 4 };
  const int bc4 = (tid & 15) * 4;

  v8f acc[2][4];
#pragma unroll
  for (int t = 0; t < 2; ++t)
#pragma unroll
    for (int nt = 0; nt < 4; ++nt) acc[t][nt] = (v8f){};

  for (int kt = 0; kt < N_; kt += TK) {
    __syncthreads();                                // prev chunk's LDS reads done
    // A tile: 256x32, global -> LDS (regs live only inside this phase)
#pragma unroll
    for (int j = 0; j < 8; ++j) {
      float4 a = *(const float4*)&adj[(rowBlk + ar[j]) * N_ + kt + ac4];
      *(float4*)&As[ar[j] * AS_STRIDE + ac4] = a;
    }
    // B tile: 32x64, global -> pair-packed LDS
#pragma unroll
    for (int j = 0; j < 2; ++j) {
      int r = br[j];
      float4 bv = *(const float4*)&U[(kt + r) * C_ + colBlk + bc4];
      float2* d = &Bp[(r >> 1) * BP_STRIDE + bc4];
      if (r & 1) { d[0].y = bv.x; d[1].y = bv.y; d[2].y = bv.z; d[3].y = bv.w; }
      else       { d[0].x = bv.x; d[1].x = bv.y; d[2].x = bv.z; d[3].x = bv.w; }
    }
    // prefetch next chunk into near caches (global_prefetch_b8; no VGPR cost)
    if (kt + TK < N_) {
#pragma unroll
      for (int j = 0; j < 8; ++j)
        __builtin_prefetch(&adj[(rowBlk + ar[j]) * N_ + (kt + TK) + ac4], 0, 3);
#pragma unroll
      for (int j = 0; j < 2; ++j)
        __builtin_prefetch(&U[(kt + TK + br[j]) * C_ + colBlk + bc4], 0, 3);
    }
    __syncthreads();                                // tiles visible

#pragma unroll
    for (int kk = 0; kk < TK; kk += 4) {
      // A frags: lane holds M=lane%16, K=kk+2*half+{0,1} -> adjacent -> one b64
      const int ac = kk + 2 * half;
      v2f a0 = *(const v2f*)&As[(rowBase +      m16) * AS_STRIDE + ac];
      v2f a1 = *(const v2f*)&As[(rowBase + 16 + m16) * AS_STRIDE + ac];
      // B frags: pair (K=kr,kr+1) pre-packed -> one b64 each
      const int kp = (kk >> 1) + half;              // kr = kk + 2*half (even)
      v2f b0 = *(const v2f*)&Bp[kp * BP_STRIDE +  0 + m16];
      v2f b1 = *(const v2f*)&Bp[kp * BP_STRIDE + 16 + m16];
      v2f b2 = *(const v2f*)&Bp[kp * BP_STRIDE + 32 + m16];
      v2f b3 = *(const v2f*)&Bp[kp * BP_STRIDE + 48 + m16];
      acc[0][0] = __builtin_amdgcn_wmma_f32_16x16x4_f32(false, a0, false, b0, (short)0, acc[0][0], false, false);
      acc[0][1] = __builtin_amdgcn_wmma_f32_16x16x4_f32(false, a0, false, b1, (short)0, acc[0][1], false, false);
      acc[0][2] = __builtin_amdgcn_wmma_f32_16x16x4_f32(false, a0, false, b2, (short)0, acc[0][2], false, false);
      acc[0][3] = __builtin_amdgcn_wmma_f32_16x16x4_f32(false, a0, false, b3, (short)0, acc[0][3], false, false);
      acc[1][0] = __builtin_amdgcn_wmma_f32_16x16x4_f32(false, a1, false, b0, (short)0, acc[1][0], false, false);
      acc[1][1] = __builtin_amdgcn_wmma_f32_16x16x4_f32(false, a1, false, b1, (short)0, acc[1][1], false, false);
      acc[1][2] = __builtin_amdgcn_wmma_f32_16x16x4_f32(false, a1, false, b2, (short)0, acc[1][2], false, false);
      acc[1][3] = __builtin_amdgcn_wmma_f32_16x16x4_f32(false, a1, false, b3, (short)0, acc[1][3], false, false);
    }
  }

  // D layout: VGPR r -> M = 8*half + r, N = lane%16
#pragma unroll
  for (int t = 0; t < 2; ++t) {
    const int outRow0 = rowBlk + rowBase + 16 * t + half * 8;
#pragma unroll
    for (int r = 0; r < 8; ++r) {
      float* dst = P + (outRow0 + r) * C_ + colBlk + m16;
      dst[0]  = acc[t][0][r];
      dst[16] = acc[t][1][r];
      dst[32] = acc[t][2][r];
      dst[48] = acc[t][3][r];
    }
  }
}

// ---------------- node update: z, Wo, ReLU, LayerNorm ----------------

__global__ __launch_bounds__(64) void node_kern(const float* __restrict__ P,
    const float* __restrict__ v, const float* __restrict__ alpha,
    const float* __restrict__ dinv, const float* __restrict__ Wo,
    const float* __restrict__ bo, const float* __restrict__ gamma,
    const float* __restrict__ beta, float* __restrict__ updated) {
  __shared__ float zsh[64];
  __shared__ float red[64];
  int bn = blockIdx.x;               // 0..B*N-1
  int b = bn / N_, n = bn % N_;
  int g = threadIdx.x;
  float di = dinv[b * N_ + n];
  float al = alpha[b * N_ + n];
  float z = di * al * P[n * C_ + b * G_ + g] + di * di * v[n * G_ + g];
  zsh[g] = z; __syncthreads();
  float y = bo[g];
  const float* w = Wo + g * G_;
#pragma unroll
  for (int gg = 0; gg < G_; ++gg) y += w[gg] * zsh[gg];
  y = fmaxf(y, 0.f);
  red[g] = y; __syncthreads();
  for (int off = 32; off > 0; off >>= 1) {
    if (g < off) red[g] += red[g + off];
    __syncthreads();
  }
  float mu = red[0] * (1.0f / 64.0f); __syncthreads();
  float dy = y - mu;
  red[g] = dy * dy; __syncthreads();
  for (int off = 32; off > 0; off >>= 1) {
    if (g < off) red[g] += red[g + off];
    __syncthreads();
  }
  float var = red[0] * (1.0f / 64.0f);
  updated[bn * G_ + g] = dy * rsqrtf(var + 1e-5f) * gamma[g] + beta[g];
}

__global__ void ps_kern(const float* __restrict__ alpha, const float* __restrict__ updated,
                        float* __restrict__ ps) {
  __shared__ float red[256];
  int b = blockIdx.x >> 6, g = blockIdx.x & 63;
  float s = 0.f;
  for (int n = threadIdx.x; n < N_; n += 256)
    s += alpha[b * N_ + n] * updated[(b * N_ + n) * G_ + g];
  red[threadIdx.x] = s; __syncthreads();
  for (int off = 128; off > 0; off >>= 1) {
    if (threadIdx.x < off) red[threadIdx.x] += red[threadIdx.x + off];
    __syncthreads();
  }
  if (threadIdx.x == 0) ps[b * G_ + g] = red[0];
}

// ---------------- launch ----------------

extern "C" void kernel_launch(void* const* d_in, const int* in_sizes, int n_in,
                              void* d_out, int out_size, void* d_ws, size_t ws_size,
                              hipStream_t stream) {
  const float* h_t       = (const float*)d_in[0];
  const float* node_repr = (const float*)d_in[1];
  const float* base_adj  = (const float*)d_in[2];
  const float* Wq        = (const float*)d_in[3];
  const float* bq        = (const float*)d_in[4];
  const float* Wk        = (const float*)d_in[5];
  const float* bk        = (const float*)d_in[6];
  const float* Wv        = (const float*)d_in[7];
  const float* bv        = (const float*)d_in[8];
  const float* Wo        = (const float*)d_in[9];
  const float* bo        = (const float*)d_in[10];
  const float* gamma     = (const float*)d_in[11];
  const float* beta      = (const float*)d_in[12];

  float* out     = (float*)d_out;
  float* ps      = out;                 // [B,G]
  float* rv      = out + B_ * G_;       // [B,G]
  float* updated = out + 2 * B_ * G_;   // [B,N,G]

  float* w = (float*)d_ws;
  float* q      = w; w += B_ * G_;
  float* k      = w; w += N_ * G_;
  float* v      = w; w += N_ * G_;
  float* rowsum = w; w += N_;
  float* logits = w; w += B_ * N_;
  float* alpha  = w; w += B_ * N_;
  float* dinv   = w; w += B_ * N_;
  float* rowmax = w; w += B_;
  float* rowexp = w; w += B_;
  float* U      = w; w += N_ * C_;
  float* P      = w; w += N_ * C_;

  q_kern      <<<(B_ * G_ + 255) / 256, 256, 0, stream>>>(h_t, Wq, bq, q);
  kv_kern     <<<(N_ * G_) / 256,       256, 0, stream>>>(node_repr, Wk, bk, Wv, bv, k, v);
  rowsum_kern <<<N_,                    256, 0, stream>>>(base_adj, rowsum);
  logits_kern <<<(B_ * N_) / 256,       256, 0, stream>>>(q, k, logits);
  smstats_kern<<<B_,                    256, 0, stream>>>(logits, rowmax, rowexp);
  alpha_kern  <<<(B_ * N_) / 256,       256, 0, stream>>>(logits, rowmax, rowexp, rowsum, alpha, dinv);
  u_kern      <<<(N_ * C_) / 256,       256, 0, stream>>>(dinv, v, U);
  rv_kern     <<<B_ * G_,               256, 0, stream>>>(alpha, v, rv);
  gemm_adj_u  <<<dim3(C_ / TN, N_ / TM), 256, 0, stream>>>(base_adj, U, P);
  node_kern   <<<B_ * N_,               64,  0, stream>>>(P, v, alpha, dinv, Wo, bo, gamma, beta, updated);
  ps_kern     <<<B_ * G_,               256, 0, stream>>>(alpha, updated, ps);
}